// TorchNeighborList_455266533928
// MI455X (gfx1250) — compile-verified
//
#include <hip/hip_runtime.h>

// CDNA5 (gfx1250) wave32 WMMA types
typedef __attribute__((ext_vector_type(2))) float v2f;  // A/B fragment: 16x4 / 4x16 f32 = 2 VGPRs
typedef __attribute__((ext_vector_type(8))) float v8f;  // C/D fragment: 16x16 f32 = 8 VGPRs

#define NB      8
#define NATOMS  2048
#define CUT2    25.0f   // CUTOFF^2 = 5.0^2

#define TI 16                 // rows per wave tile (one WMMA M)
#define TJ 64                 // cols per wave strip (4 WMMA N-subtiles)
#define WAVES_PER_BLOCK 8     // 256 threads, 8 wave32s

__global__ __launch_bounds__(256)
void neighbor_d2_wmma(const float* __restrict__ pos,
                      float* __restrict__ out_d2,
                      float* __restrict__ out_mask)
{
    const int lane      = threadIdx.x & 31;
    const int waveInBlk = threadIdx.x >> 5;
    const int gwave     = blockIdx.x * WAVES_PER_BLOCK + waveInBlk;

    const int stripsPerRow   = NATOMS / TJ;                    // 32
    const int stripsPerBatch = (NATOMS / TI) * stripsPerRow;   // 128*32 = 4096

    const int b  = gwave / stripsPerBatch;
    const int s  = gwave % stripsPerBatch;
    const int i0 = (s / stripsPerRow) * TI;
    const int j0 = (s % stripsPerRow) * TJ;

    const int m  = lane & 15;   // row/col index within 16-wide tile for this lane
    const int hi = lane >> 4;   // half-wave: selects K=0,1 (lo) vs K=2,3 (hi)

    const float* posB = pos + (size_t)b * NATOMS * 3;

    // ---- A fragment: rows i0..i0+15, K = {x,y,z,0} ----
    // ISA layout (32-bit A 16x4): lanes 0-15 -> VGPR0:K=0, VGPR1:K=1;
    //                             lanes 16-31 -> VGPR0:K=2, VGPR1:K=3.
    const float* rowA = posB + (size_t)(i0 + m) * 3;
    const float ax = rowA[0], ay = rowA[1], az = rowA[2];
    const float ni = ax * ax + ay * ay + az * az;   // |p_i|^2 for row i0+m (same in both halves)
    v2f afrag;
    afrag.x = hi ? az : ax;
    afrag.y = hi ? 0.0f : ay;

    // ---- 4 B fragments along j, 4 WMMAs: G[m][n] = p_i . p_j ----
    float nj[4];
    v8f  acc[4];
#pragma unroll
    for (int t = 0; t < 4; ++t) {
        const float* rowB = posB + (size_t)(j0 + t * 16 + m) * 3;
        const float bx = rowB[0], by = rowB[1], bz = rowB[2];
        nj[t] = bx * bx + by * by + bz * bz;        // |p_j|^2 for col j0+t*16+m
        v2f bfrag;
        bfrag.x = hi ? bz : bx;
        bfrag.y = hi ? 0.0f : by;
        v8f c = {};
        // 8 args: (neg_a, A, neg_b, B, c_mod, C, reuse_a, reuse_b)
        acc[t] = __builtin_amdgcn_wmma_f32_16x16x4_f32(
            /*neg_a=*/false, afrag, /*neg_b=*/false, bfrag,
            /*c_mod=*/(short)0, c, /*reuse_a=*/false, /*reuse_b=*/false);
    }

    // ---- Epilogue: d2 = |p_i|^2 + |p_j|^2 - 2*G, cutoff + self-pair mask ----
    // C/D layout: lane n = lane&15, row m = vgpr + 8*(lane>>4).
    const size_t baseB = (size_t)b * NATOMS * NATOMS;
#pragma unroll
    for (int v = 0; v < 8; ++v) {
        const int   rowOut = i0 + v + 8 * hi;
        const float nim    = __shfl(ni, v + 8 * hi, 32);  // gather |p_i|^2 for this output row
#pragma unroll
        for (int t = 0; t < 4; ++t) {
            const int   colOut = j0 + t * 16 + m;
            const float g  = acc[t][v];
            const float d2 = nim + nj[t] - 2.0f * g;
            const bool  valid = (d2 < CUT2) && (rowOut != colOut);
            const size_t idx = baseB + (size_t)rowOut * NATOMS + colOut;
            out_d2[idx]   = valid ? d2  : 0.0f;
            out_mask[idx] = valid ? 1.0f : 0.0f;
        }
    }
}

extern "C" void kernel_launch(void* const* d_in, const int* in_sizes, int n_in,
                              void* d_out, int out_size, void* d_ws, size_t ws_size,
                              hipStream_t stream)
{
    (void)in_sizes; (void)n_in; (void)out_size; (void)d_ws; (void)ws_size;

    const float* pos   = (const float*)d_in[0];
    float* out_d2      = (float*)d_out;
    float* out_mask    = out_d2 + (size_t)NB * NATOMS * NATOMS;

    const int totalWaves = NB * (NATOMS / TI) * (NATOMS / TJ);   // 8*128*32 = 32768
    const int blocks     = totalWaves / WAVES_PER_BLOCK;         // 4096

    neighbor_d2_wmma<<<dim3(blocks), dim3(256), 0, stream>>>(pos, out_d2, out_mask);
}